// Normalizer_33243046871372
// MI455X (gfx1250) — compile-verified
//
#include <hip/hip_runtime.h>

// Row-wise transition normalization: out = (1/deg) * adj, deg = row sum.
// Bandwidth-bound: 288 MB @ 23.3 TB/s ~= 12.4 us floor. One block per row;
// CDNA5 async global->LDS staging (ASYNCcnt path) so each input byte crosses
// HBM exactly once, NT hints on both streams to spare the 192 MB L2.

typedef float v4f __attribute__((ext_vector_type(4)));

#define NCOLS 3000
#define NBATCH 4
#define NV4 (NCOLS / 4)   // 750 float4 chunks per row (exact)
#define TPB 256           // 8 wave32 per block
#define NWAVES (TPB / 32)
#define CHUNKS 3          // ceil(750/256)

extern "C" __global__ void __launch_bounds__(TPB)
row_normalize_async(const float* __restrict__ adj, float* __restrict__ out)
{
    extern __shared__ float smem[];
    float* rowbuf = smem;          // NCOLS floats (12 KB)
    float* red    = smem + NCOLS;  // NWAVES floats

    const int row  = blockIdx.x;   // 0 .. NBATCH*NCOLS-1 (flattened [B, N])
    const int tid  = threadIdx.x;
    const int lane = tid & 31;     // wave32
    const int wid  = tid >> 5;

    // Uniform 64-bit global base of this row (GVS-mode SADDR).
    const unsigned long long rowBase =
        (unsigned long long)adj + (unsigned long long)row * (NCOLS * sizeof(float));
    // LDS byte address of rowbuf: low 32 bits of the generic pointer are the
    // LDS offset per the CDNA5 aperture mapping (addr[31:0]).
    const unsigned ldsBase = (unsigned)(size_t)rowbuf;

    // ---- Stage the whole row into LDS with async global->LDS B128 copies.
    // Consecutive lanes -> consecutive 16B: fully coalesced. NT load hint:
    // the reuse is served from LDS, so don't let the stream occupy L2.
#pragma unroll
    for (int c = 0; c < CHUNKS; ++c) {
        const int idx = tid + c * TPB;            // float4 chunk index in row
        if (idx < NV4) {
            const unsigned lOff = ldsBase + (unsigned)(idx * 16);
            const unsigned gOff = (unsigned)(idx * 16);
            asm volatile("global_load_async_to_lds_b128 %0, %1, %2 offset:0 th:TH_LOAD_NT"
                         :: "v"(lOff), "v"(gOff), "s"(rowBase)
                         : "memory");
        }
    }
    // Drain this wave's async copies, then make LDS visible workgroup-wide.
    asm volatile("s_wait_asynccnt 0" ::: "memory");
    __syncthreads();

    // ---- Pass 1: load chunks from LDS into registers, accumulate partial sum.
    const v4f* rowv4 = reinterpret_cast<const v4f*>(rowbuf);
    v4f r[CHUNKS];
    float partial = 0.0f;
#pragma unroll
    for (int c = 0; c < CHUNKS; ++c) {
        const int idx = tid + c * TPB;
        if (idx < NV4) {
            r[c] = rowv4[idx];
            partial += (r[c].x + r[c].y) + (r[c].z + r[c].w);
        }
    }

    // ---- Wave32 butterfly reduction (DS-permute hardware, no LDS traffic).
#pragma unroll
    for (int m = 16; m >= 1; m >>= 1)
        partial += __shfl_xor(partial, m, 32);

    // ---- Cross-wave: 8 partials through LDS, single barrier, broadcast read.
    if (lane == 0) red[wid] = partial;
    __syncthreads();
    float degree = 0.0f;
#pragma unroll
    for (int w = 0; w < NWAVES; ++w)
        degree += red[w];                        // broadcast reads: conflict-free

    // Reference semantics: where(deg == 0, 0, 1/deg).
    const float inv = (degree == 0.0f) ? 0.0f : (1.0f / degree);

    // ---- Pass 2: scale register-resident chunks, stream out non-temporal B128.
    v4f* outv4 = reinterpret_cast<v4f*>(out + (size_t)row * NCOLS);
#pragma unroll
    for (int c = 0; c < CHUNKS; ++c) {
        const int idx = tid + c * TPB;
        if (idx < NV4) {
            v4f v = r[c];
            v.x *= inv; v.y *= inv; v.z *= inv; v.w *= inv;
            __builtin_nontemporal_store(v, &outv4[idx]);
        }
    }
}

extern "C" void kernel_launch(void* const* d_in, const int* in_sizes, int n_in,
                              void* d_out, int out_size, void* d_ws, size_t ws_size,
                              hipStream_t stream)
{
    (void)in_sizes; (void)n_in; (void)d_ws; (void)ws_size; (void)out_size;
    const float* adj = (const float*)d_in[0];
    float* out = (float*)d_out;

    const dim3 grid(NBATCH * NCOLS);   // one block per row: 12000 blocks
    const dim3 block(TPB);
    const size_t shmem = (NCOLS + NWAVES) * sizeof(float);

    row_normalize_async<<<grid, block, shmem, stream>>>(adj, out);
}